// Model_45878840656170
// MI455X (gfx1250) — compile-verified
//
#include <hip/hip_runtime.h>
#include <hip/hip_bf16.h>
#include <math.h>

// ---------------------------------------------------------------------------
// Types for WMMA fragments (fp32 path: V_WMMA_F32_16X16X4_F32)
// ---------------------------------------------------------------------------
typedef __attribute__((ext_vector_type(2))) float f32x2;
typedef __attribute__((ext_vector_type(8))) float f32x8;

#define D_DIM 128

// ---------------------------------------------------------------------------
// Initial features: xs = x @ lin_w + lin_b + emb[node_id]
// blockDim = 128 (one block per node), gridDim = N
// ---------------------------------------------------------------------------
__global__ void init_features(const float* __restrict__ x,
                              const float* __restrict__ lw,
                              const float* __restrict__ lb,
                              const float* __restrict__ emb,
                              const int*   __restrict__ nid,
                              float* __restrict__ xs)
{
    __shared__ float xrow[20];
    const int n = blockIdx.x;
    const int d = threadIdx.x;
    if (threadIdx.x < 20) xrow[threadIdx.x] = x[(size_t)n * 20 + threadIdx.x];
    __syncthreads();
    float acc = lb[d];
#pragma unroll
    for (int k = 0; k < 20; ++k) acc += xrow[k] * lw[k * D_DIM + d];
    acc += emb[(size_t)nid[n] * D_DIM + d];
    xs[(size_t)n * D_DIM + d] = acc;
}

// ---------------------------------------------------------------------------
// Y[N,128] = X[N,128] @ W (+ bias).  TB==1 -> Y = X @ W^T.
// Wave-per-16x16-tile; fp32 WMMA 16x16x4, K-loop of 32 steps.
// blockDim = 128 (4 waves), grid = (N/16, 2)  => 8 column tiles covered.
// A layout (ISA 7.12.2, 32-bit A 16x4): lanes 0-15 hold K=0,1; lanes 16-31 K=2,3.
// C/D layout: VGPR r -> (M=r, N=lane) lanes 0-15 ; (M=r+8, N=lane-16) lanes 16-31.
// ---------------------------------------------------------------------------
template <int TB>
__global__ void gemm128_wmma(const float* __restrict__ X,
                             const float* __restrict__ W,
                             const float* __restrict__ bias,
                             float* __restrict__ Y)
{
    const int lane = threadIdx.x & 31;
    const int wave = threadIdx.x >> 5;
    const int half = lane >> 4;        // 0: lanes 0-15, 1: lanes 16-31
    const int lm   = lane & 15;
    const int mBase = blockIdx.x << 4;
    const int nBase = ((blockIdx.y << 2) + wave) << 4;
    const int row = mBase + lm;
    const int col = nBase + lm;

    const float* __restrict__ xr = X + (size_t)row * D_DIM;

    f32x8 c = {0.f, 0.f, 0.f, 0.f, 0.f, 0.f, 0.f, 0.f};

#pragma unroll 4
    for (int kk = 0; kk < D_DIM; kk += 4) {
        const int ka = kk + 2 * half;
        f32x2 a, b;
        a.x = xr[ka];
        a.y = xr[ka + 1];
        if (TB == 0) {                 // B[k, col] = W[k*128 + col]
            b.x = W[(size_t)ka * D_DIM + col];
            b.y = W[(size_t)(ka + 1) * D_DIM + col];
        } else {                       // B[k, col] = W[col*128 + k]  (W^T)
            b.x = W[(size_t)col * D_DIM + ka];
            b.y = W[(size_t)col * D_DIM + ka + 1];
        }
        // D = A(16x4) * B(4x16) + C
        c = __builtin_amdgcn_wmma_f32_16x16x4_f32(false, a, false, b,
                                                  (short)0, c, false, false);
    }

    const float bv = bias ? bias[col] : 0.f;
#pragma unroll
    for (int r = 0; r < 8; ++r) {
        const int orow = mBase + r + 8 * half;
        Y[(size_t)orow * D_DIM + col] = c[r] + bv;
    }
}

// ---------------------------------------------------------------------------
// Utility fills / relu
// ---------------------------------------------------------------------------
__global__ void fill_f32(float* __restrict__ p, float v, long long n)
{
    long long i = (long long)blockIdx.x * blockDim.x + threadIdx.x;
    long long stride = (long long)gridDim.x * blockDim.x;
    for (; i < n; i += stride) p[i] = v;
}

__global__ void relu_copy(const float* __restrict__ src, float* __restrict__ dst,
                          long long n)
{
    long long i = (long long)blockIdx.x * blockDim.x + threadIdx.x;
    long long stride = (long long)gridDim.x * blockDim.x;
    for (; i < n; i += stride) {
        float v = src[i];
        dst[i] = v > 0.f ? v : 0.f;
    }
}

// ---------------------------------------------------------------------------
// float atomic-max via CAS (works for any sign, -inf init)
// ---------------------------------------------------------------------------
__device__ inline void atomicMaxFloat(float* addr, float value)
{
    float old = *addr;
    while (old < value) {
        int assumed = __float_as_int(old);
        int prev = atomicCAS((int*)addr, assumed, __float_as_int(value));
        if (prev == assumed) break;
        old = __int_as_float(prev);
    }
}

// ---------------------------------------------------------------------------
// Edge phase. ei layout: [2, E] -> src = ei[e], dst = ei[E + e].
// Wave per edge; lane does one float4 of the 128-float rows (coalesced).
// ---------------------------------------------------------------------------
__global__ void edge_score(const float* __restrict__ Q,
                           const float* __restrict__ WK,
                           const int*   __restrict__ ei,
                           int srcBase, int dstBase, int E,
                           float* __restrict__ S,
                           float* __restrict__ M)
{
    const int wid  = blockIdx.x * (blockDim.x >> 5) + (threadIdx.x >> 5);
    const int lane = threadIdx.x & 31;
    if (wid >= E) return;
    const int sg = srcBase + ei[wid];
    const int dg = dstBase + ei[E + wid];
    const float4* q4 = (const float4*)(Q  + (size_t)sg * D_DIM);
    const float4* w4 = (const float4*)(WK + (size_t)dg * D_DIM);
    const float4 qa = q4[lane];
    const float4 wa = w4[lane];
    float p = qa.x * wa.x + qa.y * wa.y + qa.z * wa.z + qa.w * wa.w;
#pragma unroll
    for (int off = 16; off > 0; off >>= 1) p += __shfl_xor(p, off, 32);
    if (lane == 0) {
        float s = p * 0.08838834764831845f;   // 1/sqrt(128)
        s = s > 0.f ? s : 0.01f * s;          // leaky_relu, slope 0.01
        S[wid] = s;
        atomicMaxFloat(&M[dg], s);
    }
}

__global__ void edge_expsum(float* __restrict__ S,
                            const int* __restrict__ ei,
                            int dstBase, int E,
                            const float* __restrict__ M,
                            float* __restrict__ Den)
{
    const int e = blockIdx.x * blockDim.x + threadIdx.x;
    if (e >= E) return;
    const int dg = dstBase + ei[E + e];
    const float ex = expf(S[e] - M[dg]);
    S[e] = ex;
    atomicAdd(&Den[dg], ex);
}

__global__ void edge_scatter(const float* __restrict__ S,
                             const float* __restrict__ Den,
                             const float* __restrict__ V,
                             const int*   __restrict__ ei,
                             int srcBase, int dstBase, int E,
                             float* __restrict__ out)
{
    const int wid  = blockIdx.x * (blockDim.x >> 5) + (threadIdx.x >> 5);
    const int lane = threadIdx.x & 31;
    if (wid >= E) return;
    const int sg = srcBase + ei[wid];
    const int dg = dstBase + ei[E + wid];
    const float alpha = S[wid] / Den[dg];
    const float* v = V   + (size_t)sg * D_DIM;
    float*       o = out + (size_t)dg * D_DIM;
#pragma unroll
    for (int c = 0; c < 4; ++c) {
        const int idx = lane + 32 * c;
        atomicAdd(&o[idx], alpha * v[idx]);
    }
}

// ---------------------------------------------------------------------------
// Host orchestration
// ---------------------------------------------------------------------------
extern "C" void kernel_launch(void* const* d_in, const int* in_sizes, int n_in,
                              void* d_out, int out_size, void* d_ws, size_t ws_size,
                              hipStream_t stream)
{
    (void)n_in; (void)out_size; (void)ws_size;

    const float* x[3]   = {(const float*)d_in[0], (const float*)d_in[1], (const float*)d_in[2]};
    const int*   nid[3] = {(const int*)d_in[3], (const int*)d_in[4], (const int*)d_in[5]};
    const int*   ei[3]  = {(const int*)d_in[6], (const int*)d_in[7], (const int*)d_in[8]};
    const float* lin_w  = (const float*)d_in[9];
    const float* lin_b  = (const float*)d_in[10];
    const float* emb[3] = {(const float*)d_in[11], (const float*)d_in[12], (const float*)d_in[13]};
    const float* qkv_w  = (const float*)d_in[14];
    const float* qkv_b  = (const float*)d_in[15];
    const float* W_edge = (const float*)d_in[16];

    const int Nc[3] = {in_sizes[0] / 20, in_sizes[1] / 20, in_sizes[2] / 20};
    const int base[3] = {0, Nc[0], Nc[0] + Nc[1]};
    const int NT = Nc[0] + Nc[1] + Nc[2];
    const long long ND = (long long)NT * D_DIM;

    const int Ecnt[3] = {in_sizes[6] / 2, in_sizes[7] / 2, in_sizes[8] / 2};
    int maxE = Ecnt[0];
    if (Ecnt[1] > maxE) maxE = Ecnt[1];
    if (Ecnt[2] > maxE) maxE = Ecnt[2];

    // edge-type topology: (src_type, dst_type)
    const int srcT[3] = {1, 0, 2};
    const int dstT[3] = {0, 2, 1};

    // workspace layout (floats)
    float* ws  = (float*)d_ws;
    float* XS  = ws;
    float* Qb  = ws + 1 * ND;
    float* Kb  = ws + 2 * ND;   // reused as layer-0 output after WK is built
    float* Vb  = ws + 3 * ND;
    float* WKb = ws + 4 * ND;
    float* Sb  = ws + 5 * ND;
    float* Mb  = Sb + maxE;
    float* Db  = Mb + NT;

    // --- initial features -------------------------------------------------
    for (int t = 0; t < 3; ++t) {
        init_features<<<Nc[t], 128, 0, stream>>>(
            x[t], lin_w + (size_t)t * 20 * D_DIM, lin_b + (size_t)t * D_DIM,
            emb[t], nid[t], XS + (size_t)base[t] * D_DIM);
    }

    const int fillBlocks = 4096;

    for (int l = 0; l < 2; ++l) {
        // --- Q/K/V GEMMs (fp32 WMMA) -------------------------------------
        for (int t = 0; t < 3; ++t) {
            const dim3 grid(Nc[t] / 16, 2);
            const size_t wb = (size_t)((l * 3 + t) * 3);
            gemm128_wmma<0><<<grid, 128, 0, stream>>>(
                XS + (size_t)base[t] * D_DIM, qkv_w + (wb + 0) * D_DIM * D_DIM,
                qkv_b + (wb + 0) * D_DIM, Qb + (size_t)base[t] * D_DIM);
            gemm128_wmma<0><<<grid, 128, 0, stream>>>(
                XS + (size_t)base[t] * D_DIM, qkv_w + (wb + 1) * D_DIM * D_DIM,
                qkv_b + (wb + 1) * D_DIM, Kb + (size_t)base[t] * D_DIM);
            gemm128_wmma<0><<<grid, 128, 0, stream>>>(
                XS + (size_t)base[t] * D_DIM, qkv_w + (wb + 2) * D_DIM * D_DIM,
                qkv_b + (wb + 2) * D_DIM, Vb + (size_t)base[t] * D_DIM);
        }
        // --- WK = K @ W_edge^T for each edge type's destination ----------
        for (int et = 0; et < 3; ++et) {
            const int dt = dstT[et];
            const dim3 grid(Nc[dt] / 16, 2);
            gemm128_wmma<1><<<grid, 128, 0, stream>>>(
                Kb + (size_t)base[dt] * D_DIM,
                W_edge + (size_t)(l * 3 + et) * D_DIM * D_DIM,
                nullptr, WKb + (size_t)base[dt] * D_DIM);
        }

        // K is dead now; reuse it as the aggregation output for layer 0.
        float* out = (l == 0) ? Kb : (float*)d_out;
        fill_f32<<<fillBlocks, 256, 0, stream>>>(out, 0.f, ND);
        fill_f32<<<fillBlocks, 256, 0, stream>>>(Mb, -INFINITY, (long long)NT);
        fill_f32<<<fillBlocks, 256, 0, stream>>>(Db, 0.f, (long long)NT);

        // --- edge pipeline per edge type ---------------------------------
        for (int et = 0; et < 3; ++et) {
            const int E = Ecnt[et];
            const int sB = base[srcT[et]];
            const int dB = base[dstT[et]];
            const int waveBlocks = (E + 7) / 8;       // 8 waves per 256-thread block
            edge_score<<<waveBlocks, 256, 0, stream>>>(Qb, WKb, ei[et], sB, dB, E, Sb, Mb);
            edge_expsum<<<(E + 255) / 256, 256, 0, stream>>>(Sb, ei[et], dB, E, Mb, Db);
            edge_scatter<<<waveBlocks, 256, 0, stream>>>(Sb, Db, Vb, ei[et], sB, dB, E, out);
        }

        if (l == 0) {
            relu_copy<<<fillBlocks, 256, 0, stream>>>(out, XS, ND);
        }
    }
}